// GC_SS2D_79121887527136
// MI455X (gfx1250) — compile-verified
//
#include <hip/hip_runtime.h>
#include <hip/hip_bf16.h>

#if defined(__HIP_DEVICE_COMPILE__)
#if !__has_builtin(__builtin_amdgcn_wmma_f32_16x16x4_f32)
#error "wmma_f32_16x16x4_f32 builtin not available on this toolchain (device pass)"
#endif
#endif

typedef __attribute__((ext_vector_type(2))) float v2f;
typedef __attribute__((ext_vector_type(8))) float v8f;

// Problem constants (match reference)
constexpr int B_  = 2;
constexpr int C_  = 192;   // DINNER
constexpr int H_  = 48;
constexpr int W_  = 48;
constexpr int L_  = 2304;  // H*W
constexpr int K_  = 4;
constexpr int N_  = 16;
constexpr int R_  = 12;
constexpr int D44 = R_ + 2 * N_;  // 44 rows of x_dbl
constexpr int DPAD = 48;          // padded to 3 WMMA row-tiles
constexpr int LC  = 64;           // scan chunk length
constexpr int G_  = L_ / LC;      // 36 chunks
constexpr int TILES_L = L_ / 16;  // 144

// ---------------------------------------------------------------------------
// 0) pad W1 (K,44,192) -> (K,48,192) with zero rows; removes divergent
//    exec-masked loads from the WMMA inner loop
// ---------------------------------------------------------------------------
__global__ void k_pad_w1(const float* __restrict__ w1, float* __restrict__ w1p) {
  int idx = blockIdx.x * blockDim.x + threadIdx.x;
  if (idx >= K_ * DPAD * C_) return;
  int cc = idx % C_;
  int t  = idx / C_;
  int d  = t % DPAD;
  int k  = t / DPAD;
  w1p[idx] = (d < D44) ? w1[((size_t)k * D44 + d) * C_ + cc] : 0.0f;
}

// ---------------------------------------------------------------------------
// 1) cross-scan: build xs[b][k][c][l] from x[b][c][h][w]
// ---------------------------------------------------------------------------
__global__ void k_cross_scan(const float* __restrict__ x, float* __restrict__ xs) {
  int idx = blockIdx.x * blockDim.x + threadIdx.x;
  if (idx >= B_ * K_ * C_ * L_) return;
  int l = idx % L_;
  int t = idx / L_;
  int c = t % C_; t /= C_;
  int k = t % K_;
  int b = t / K_;
  const float* xr = x + (size_t)(b * C_ + c) * L_;
  float v;
  if (k == 0) {
    v = xr[l];
  } else if (k == 1) {            // transposed flatten: l = w*H + h
    int hh = l % H_, ww = l / H_;
    v = xr[hh * W_ + ww];
  } else if (k == 2) {            // reversed
    v = xr[L_ - 1 - l];
  } else {                        // transposed + reversed
    int lr = L_ - 1 - l;
    int hh = lr % H_, ww = lr / H_;
    v = xr[hh * W_ + ww];
  }
  xs[idx] = v;
}

// ---------------------------------------------------------------------------
// 2) GEMM1 (WMMA f32 16x16x4): x_dbl(48xL) = W1p(48x192) * xs(192xL), per (b,k)
//    one wave per 16x16 output tile; no divergence anywhere
// ---------------------------------------------------------------------------
__global__ void k_gemm1_wmma(const float* __restrict__ xs,
                             const float* __restrict__ w1p,  // (K,48,192) padded
                             float* __restrict__ xdbl) {     // (B*K, 48, L)
  int wave = blockIdx.x * (blockDim.x >> 5) + (threadIdx.x >> 5);
  int lane = threadIdx.x & 31;
  int t  = wave;
  int lt = t % TILES_L; t /= TILES_L;
  int mt = t % 3;       t /= 3;
  int bk = t;
  if (bk >= B_ * K_) return;
  int k  = bk % K_;
  int l0 = lt * 16, d0 = mt * 16;
  int m  = lane & 15, hi = lane >> 4;

  const float* Wk = w1p + (size_t)k * DPAD * C_;
  const float* Xs = xs + (size_t)bk * C_ * L_;
  const float* Arow = Wk + (size_t)(d0 + m) * C_ + 2 * hi;  // A row, lane K-slot base
  const float* Bcol = Xs + (size_t)(2 * hi) * L_ + l0 + m;  // B lane base

  v8f acc = {};
  for (int kk = 0; kk < C_; kk += 4) {
    v2f a = *(const v2f*)(Arow + kk);          // A[m][kk+2hi], A[m][kk+2hi+1]
    v2f bf;
    bf[0] = Bcol[(size_t)kk * L_];             // B[kk+2hi][col]
    bf[1] = Bcol[(size_t)kk * L_ + L_];        // B[kk+2hi+1][col]
    acc = __builtin_amdgcn_wmma_f32_16x16x4_f32(
        /*neg_a=*/false, a, /*neg_b=*/false, bf,
        /*c_mod=*/(short)0, acc, /*reuse_a=*/false, /*reuse_b=*/false);
  }
  float* Od = xdbl + (size_t)bk * DPAD * L_;
  int col = l0 + m;
#pragma unroll
  for (int r = 0; r < 8; ++r) {
    int row = d0 + r + hi * 8;   // C/D layout: VGPR r -> rows r / r+8
    Od[(size_t)row * L_ + col] = acc[r];
  }
}

// ---------------------------------------------------------------------------
// 3) GEMM2 (WMMA f32 16x16x4): delta(192xL) = W2(192x12) * dts(12xL), per (b,k)
// ---------------------------------------------------------------------------
__global__ void k_gemm2_wmma(const float* __restrict__ xdbl,
                             const float* __restrict__ w2,    // (K,192,12)
                             float* __restrict__ delta) {     // (B*K, C, L)
  int wave = blockIdx.x * (blockDim.x >> 5) + (threadIdx.x >> 5);
  int lane = threadIdx.x & 31;
  int t  = wave;
  int lt = t % TILES_L; t /= TILES_L;
  int mt = t % (C_ / 16); t /= (C_ / 16);
  int bk = t;
  if (bk >= B_ * K_) return;
  int k  = bk % K_;
  int l0 = lt * 16, d0 = mt * 16;
  int m  = lane & 15, hi = lane >> 4;

  const float* Wk = w2 + (size_t)k * C_ * R_;
  const float* Bm = xdbl + (size_t)bk * DPAD * L_;  // rows 0..11 = dts
  const float* Arow = Wk + (size_t)(d0 + m) * R_ + 2 * hi;
  const float* Bcol = Bm + (size_t)(2 * hi) * L_ + l0 + m;

  v8f acc = {};
#pragma unroll
  for (int kk = 0; kk < R_; kk += 4) {
    v2f a = *(const v2f*)(Arow + kk);
    v2f bf;
    bf[0] = Bcol[(size_t)kk * L_];
    bf[1] = Bcol[(size_t)kk * L_ + L_];
    acc = __builtin_amdgcn_wmma_f32_16x16x4_f32(
        false, a, false, bf, (short)0, acc, false, false);
  }
  float* Od = delta + (size_t)bk * C_ * L_;
  int col = l0 + m;
#pragma unroll
  for (int r = 0; r < 8; ++r) {
    int row = d0 + r + hi * 8;
    Od[(size_t)row * L_ + col] = acc[r];
  }
}

// ---------------------------------------------------------------------------
// Scan helpers
// ---------------------------------------------------------------------------
__device__ __forceinline__ float softplus_f(float x) {
  return (x > 20.0f) ? x : log1pf(__expf(x));
}

// ---------------------------------------------------------------------------
// 4) scan pass A: per chunk, compute P[n] = prod(dA) and S[n] = state from h=0
//    block = 192 threads (one per channel c), grid = (B*K)*G
// ---------------------------------------------------------------------------
__global__ void k_scan_partial(const float* __restrict__ xs,
                               const float* __restrict__ xdbl,
                               const float* __restrict__ delta,
                               const float* __restrict__ A_logs,
                               const float* __restrict__ dt_bias,
                               float* __restrict__ Pout,
                               float* __restrict__ Sout) {
  __shared__ float sB[N_][LC];
  int g  = blockIdx.x % G_;
  int bk = blockIdx.x / G_;
  int c  = threadIdx.x;
  int k  = bk % K_;

  for (int idx = c; idx < N_ * LC; idx += C_) {
    int n = idx / LC, col = idx % LC;
    sB[n][col] = xdbl[((size_t)bk * DPAD + R_ + n) * L_ + g * LC + col];
  }
  __syncthreads();

  int kc = k * C_ + c;
  float Arow[N_];
#pragma unroll
  for (int n = 0; n < N_; ++n) Arow[n] = -__expf(A_logs[(size_t)kc * N_ + n]);
  float bias = dt_bias[kc];

  const float* up = xs    + ((size_t)bk * C_ + c) * L_ + g * LC;
  const float* dp = delta + ((size_t)bk * C_ + c) * L_ + g * LC;

  float h[N_], P[N_];
#pragma unroll
  for (int n = 0; n < N_; ++n) { h[n] = 0.0f; P[n] = 1.0f; }

  for (int ll = 0; ll < LC; ll += 4) {
    float4 u4 = *(const float4*)(up + ll);
    float4 d4 = *(const float4*)(dp + ll);
    float uu[4] = {u4.x, u4.y, u4.z, u4.w};
    float dd[4] = {d4.x, d4.y, d4.z, d4.w};
#pragma unroll
    for (int j = 0; j < 4; ++j) {
      float dlt = softplus_f(dd[j] + bias);
      float du  = dlt * uu[j];
#pragma unroll
      for (int n = 0; n < N_; ++n) {
        float dA = __expf(dlt * Arow[n]);
        h[n] = dA * h[n] + du * sB[n][ll + j];
        P[n] *= dA;
      }
    }
  }
  size_t base = ((size_t)(bk * C_ + c) * N_) * G_ + g;
#pragma unroll
  for (int n = 0; n < N_; ++n) {
    Pout[base + (size_t)n * G_] = P[n];
    Sout[base + (size_t)n * G_] = h[n];
  }
}

// ---------------------------------------------------------------------------
// 5) scan pass B: sequential fix-up across chunks (per b,k,c,n), 36 steps
// ---------------------------------------------------------------------------
__global__ void k_scan_fixup(const float* __restrict__ P,
                             const float* __restrict__ S,
                             float* __restrict__ Hs) {
  int idx = blockIdx.x * blockDim.x + threadIdx.x;
  if (idx >= B_ * K_ * C_ * N_) return;
  const float* Pp = P + (size_t)idx * G_;
  const float* Sp = S + (size_t)idx * G_;
  float* Hp = Hs + (size_t)idx * G_;
  float h = 0.0f;
  for (int g = 0; g < G_; ++g) {
    Hp[g] = h;                 // state entering chunk g
    h = Pp[g] * h + Sp[g];
  }
}

// ---------------------------------------------------------------------------
// 6) scan pass C: replay chunk with correct initial state, emit y
// ---------------------------------------------------------------------------
__global__ void k_scan_final(const float* __restrict__ xs,
                             const float* __restrict__ xdbl,
                             const float* __restrict__ delta,
                             const float* __restrict__ A_logs,
                             const float* __restrict__ dt_bias,
                             const float* __restrict__ Ds,
                             const float* __restrict__ Hs,
                             float* __restrict__ y) {
  __shared__ float sB[N_][LC];
  __shared__ float sC[N_][LC];
  int g  = blockIdx.x % G_;
  int bk = blockIdx.x / G_;
  int c  = threadIdx.x;
  int k  = bk % K_;

  for (int idx = c; idx < 2 * N_ * LC; idx += C_) {
    int n = idx / LC, col = idx % LC;
    float v = xdbl[((size_t)bk * DPAD + R_ + n) * L_ + g * LC + col];
    if (n < N_) sB[n][col] = v;
    else        sC[n - N_][col] = v;
  }
  __syncthreads();

  int kc = k * C_ + c;
  float Arow[N_];
#pragma unroll
  for (int n = 0; n < N_; ++n) Arow[n] = -__expf(A_logs[(size_t)kc * N_ + n]);
  float bias = dt_bias[kc];
  float Dv = Ds[kc];

  const float* up = xs    + ((size_t)bk * C_ + c) * L_ + g * LC;
  const float* dp = delta + ((size_t)bk * C_ + c) * L_ + g * LC;
  float* yp       = y     + ((size_t)bk * C_ + c) * L_ + g * LC;

  float h[N_];
  size_t hbase = ((size_t)(bk * C_ + c) * N_) * G_ + g;
#pragma unroll
  for (int n = 0; n < N_; ++n) h[n] = Hs[hbase + (size_t)n * G_];

  for (int ll = 0; ll < LC; ll += 4) {
    float4 u4 = *(const float4*)(up + ll);
    float4 d4 = *(const float4*)(dp + ll);
    float uu[4] = {u4.x, u4.y, u4.z, u4.w};
    float dd[4] = {d4.x, d4.y, d4.z, d4.w};
#pragma unroll
    for (int j = 0; j < 4; ++j) {
      float dlt = softplus_f(dd[j] + bias);
      float du  = dlt * uu[j];
      float acc = 0.0f;
#pragma unroll
      for (int n = 0; n < N_; ++n) {
        float dA = __expf(dlt * Arow[n]);
        h[n] = dA * h[n] + du * sB[n][ll + j];
        acc += h[n] * sC[n][ll + j];
      }
      yp[ll + j] = acc + uu[j] * Dv;
    }
  }
}

// ---------------------------------------------------------------------------
// 7) cross-merge: out[b,c,h,w] = y0[l] + y2[L-1-l] + y1[lt] + y3[L-1-lt]
// ---------------------------------------------------------------------------
__global__ void k_cross_merge(const float* __restrict__ y, float* __restrict__ out) {
  int idx = blockIdx.x * blockDim.x + threadIdx.x;
  if (idx >= B_ * C_ * H_ * W_) return;
  int w = idx % W_;
  int t = idx / W_;
  int hh = t % H_; t /= H_;
  int c = t % C_;
  int b = t / C_;
  int l  = hh * W_ + w;
  int lt = w * H_ + hh;
  const float* yb = y + (size_t)b * K_ * C_ * L_;
  float r = yb[((size_t)0 * C_ + c) * L_ + l]
          + yb[((size_t)2 * C_ + c) * L_ + (L_ - 1 - l)]
          + yb[((size_t)1 * C_ + c) * L_ + lt]
          + yb[((size_t)3 * C_ + c) * L_ + (L_ - 1 - lt)];
  out[idx] = r;
}

// ---------------------------------------------------------------------------
// launch
// ---------------------------------------------------------------------------
extern "C" void kernel_launch(void* const* d_in, const int* in_sizes, int n_in,
                              void* d_out, int out_size, void* d_ws, size_t ws_size,
                              hipStream_t stream) {
  const float* x        = (const float*)d_in[0];  // (B,C,H,W)
  const float* w1       = (const float*)d_in[1];  // (K,44,192)
  const float* w2       = (const float*)d_in[2];  // (K,192,12)
  const float* A_logs   = (const float*)d_in[3];  // (K*C, N)
  const float* Ds       = (const float*)d_in[4];  // (K*C)
  const float* dt_bias  = (const float*)d_in[5];  // (K*C)
  float* out = (float*)d_out;                     // (B,C,H,W)

  const size_t NXS = (size_t)B_ * K_ * C_ * L_;       // 3,538,944
  const size_t NXD = (size_t)B_ * K_ * DPAD * L_;     //   884,736
  const size_t NPS = (size_t)B_ * K_ * C_ * N_ * G_;  //   884,736
  const size_t NW1 = (size_t)K_ * DPAD * C_;          //    36,864

  float* ws    = (float*)d_ws;
  float* xs    = ws;
  float* xdbl  = xs + NXS;
  float* delta = xdbl + NXD;
  float* yscan = delta + NXS;
  float* Pbuf  = yscan + NXS;
  float* Sbuf  = Pbuf + NPS;
  float* Hbuf  = Sbuf + NPS;
  float* w1pad = Hbuf + NPS;

  // 0) zero-pad W1 rows 44..47 (one-shot, tiny)
  {
    int n = K_ * DPAD * C_;
    k_pad_w1<<<(n + 255) / 256, 256, 0, stream>>>(w1, w1pad);
  }
  // 1) cross-scan
  {
    int n = B_ * K_ * C_ * L_;
    k_cross_scan<<<(n + 255) / 256, 256, 0, stream>>>(x, xs);
  }
  // 2) GEMM1 via WMMA f32 (one wave per 16x16 tile)
  {
    int waves = B_ * K_ * 3 * TILES_L;     // 3456
    k_gemm1_wmma<<<waves / 4, 128, 0, stream>>>(xs, w1pad, xdbl);
  }
  // 3) GEMM2 via WMMA f32
  {
    int waves = B_ * K_ * (C_ / 16) * TILES_L;  // 13824
    k_gemm2_wmma<<<waves / 4, 128, 0, stream>>>(xdbl, w2, delta);
  }
  // 4) chunked scan: partial
  k_scan_partial<<<B_ * K_ * G_, C_, 0, stream>>>(xs, xdbl, delta, A_logs, dt_bias,
                                                  Pbuf, Sbuf);
  // 5) chunk fix-up
  {
    int n = B_ * K_ * C_ * N_;
    k_scan_fixup<<<(n + 255) / 256, 256, 0, stream>>>(Pbuf, Sbuf, Hbuf);
  }
  // 6) final scan + y
  k_scan_final<<<B_ * K_ * G_, C_, 0, stream>>>(xs, xdbl, delta, A_logs, dt_bias,
                                                Ds, Hbuf, yscan);
  // 7) cross-merge
  {
    int n = B_ * C_ * H_ * W_;
    k_cross_merge<<<(n + 255) / 256, 256, 0, stream>>>(yscan, out);
  }
  (void)in_sizes; (void)n_in; (void)out_size; (void)ws_size;
}